// Model_61314953118506
// MI455X (gfx1250) — compile-verified
//
#include <hip/hip_runtime.h>
#include <hip/hip_bf16.h>
#include <math.h>

typedef __attribute__((ext_vector_type(2))) float v2f;
typedef __attribute__((ext_vector_type(8))) float v8f;

#define DD 64

// ---------------------------------------------------------------------------
// zero fill
// ---------------------------------------------------------------------------
__global__ void zerof_k(float* __restrict__ p, int n) {
    int i = blockIdx.x * blockDim.x + threadIdx.x;
    int st = gridDim.x * blockDim.x;
    for (; i < n; i += st) p[i] = 0.0f;
}

__global__ void zeroi_k(int* __restrict__ p, int n) {
    int i = blockIdx.x * blockDim.x + threadIdx.x;
    int st = gridDim.x * blockDim.x;
    for (; i < n; i += st) p[i] = 0;
}

// ---------------------------------------------------------------------------
// counting sort of edges by relation
// ---------------------------------------------------------------------------
__global__ void hist_k(const int* __restrict__ etype, int* __restrict__ counts, int E) {
    int i = blockIdx.x * blockDim.x + threadIdx.x;
    int st = gridDim.x * blockDim.x;
    for (; i < E; i += st) atomicAdd(&counts[etype[i]], 1);
}

__global__ void scan_k(const int* __restrict__ counts, int* __restrict__ offsets,
                       int* __restrict__ cursor, int R) {
    if (blockIdx.x == 0 && threadIdx.x == 0) {
        int s = 0;
        for (int r = 0; r < R; ++r) { offsets[r] = s; cursor[r] = s; s += counts[r]; }
        offsets[R] = s;
    }
}

__global__ void scatter_k(const int* __restrict__ src, const int* __restrict__ dst,
                          const int* __restrict__ etype, int* __restrict__ cursor,
                          int* __restrict__ srcS, int* __restrict__ dstS, int E) {
    int i = blockIdx.x * blockDim.x + threadIdx.x;
    int st = gridDim.x * blockDim.x;
    for (; i < E; i += st) {
        int r = etype[i];
        int slot = atomicAdd(&cursor[r], 1);
        srcS[slot] = src[i];
        dstS[slot] = dst[i];
    }
}

// ---------------------------------------------------------------------------
// T[n][o] = sum_k x[n][k] * W[k][o]   (x: [nrows,64], W: [64,64] row-major)
// One wave32 per 16-row tile. FP32 WMMA 16x16x4, K accumulated over 16 steps,
// 4 output column tiles -> 64 v_wmma_f32_16x16x4_f32 per wave.
//
// A (16x4 f32, 2 VGPR): lanes 0-15 hold M=lane, K={0,1}; lanes 16-31 hold
//   M=lane-16, K={2,3}  (ISA 7.12.2 "32-bit A-Matrix 16x4").
// B (4x16 f32, 2 VGPR): mirrored: lanes 0-15 rows K={0,1} (N=lane),
//   lanes 16-31 rows K={2,3}.
// C/D (16x16 f32, 8 VGPR): VGPR v: lanes 0-15 -> (M=v, N=lane),
//   lanes 16-31 -> (M=v+8, N=lane-16).
// ---------------------------------------------------------------------------
__global__ void transform_k(const float* __restrict__ x, const float* __restrict__ W,
                            float* __restrict__ T, int nrows) {
    int wave = (blockIdx.x * blockDim.x + threadIdx.x) >> 5;
    int lane = threadIdx.x & 31;
    int row0 = wave * 16;
    if (row0 >= nrows) return;           // wave-uniform: EXEC all-ones at WMMA

    int l15 = lane & 15;
    int hi  = lane >> 4;                 // 0: K-pair {0,1}; 1: K-pair {2,3}

    const float* xrow = x + (long)(row0 + l15) * DD;

    // preload all 16 A k-tiles for this 16x64 row block
    v2f a[16];
#pragma unroll
    for (int kt = 0; kt < 16; ++kt) {
        a[kt] = *(const v2f*)(xrow + 4 * kt + 2 * hi);
    }

    v8f c0 = {}, c1 = {}, c2 = {}, c3 = {};
#pragma unroll
    for (int kt = 0; kt < 16; ++kt) {
        int krow = 4 * kt + 2 * hi;
        const float* wp = W + krow * DD + l15;   // W[krow][l15 + 16*j]
        v2f b;
        b.x = wp[0];       b.y = wp[DD];
        c0 = __builtin_amdgcn_wmma_f32_16x16x4_f32(false, a[kt], false, b, (short)0, c0, false, false);
        b.x = wp[16];      b.y = wp[16 + DD];
        c1 = __builtin_amdgcn_wmma_f32_16x16x4_f32(false, a[kt], false, b, (short)0, c1, false, false);
        b.x = wp[32];      b.y = wp[32 + DD];
        c2 = __builtin_amdgcn_wmma_f32_16x16x4_f32(false, a[kt], false, b, (short)0, c2, false, false);
        b.x = wp[48];      b.y = wp[48 + DD];
        c3 = __builtin_amdgcn_wmma_f32_16x16x4_f32(false, a[kt], false, b, (short)0, c3, false, false);
    }

    float* out = T + (long)row0 * DD;
#pragma unroll
    for (int v = 0; v < 8; ++v) {
        int M = v + 8 * hi;
        float* orow = out + (long)M * DD + l15;
        orow[0]  = c0[v];
        orow[16] = c1[v];
        orow[32] = c2[v];
        orow[48] = c3[v];
    }
}

// ---------------------------------------------------------------------------
// per-relation edge aggregation: for each edge of relation r,
//   agg[dst] += T_r[src]   (64 floats, f32 global atomics; 4 threads/edge)
// Grid-stride over this relation's contiguous sorted-edge range.
// ---------------------------------------------------------------------------
__global__ void edge_agg_k(const float* __restrict__ T, const int* __restrict__ srcS,
                           const int* __restrict__ dstS, const int* __restrict__ offsets,
                           int r, float* __restrict__ agg) {
    int base = offsets[r];
    int cnt  = offsets[r + 1] - base;
    long total = (long)cnt * 4;
    long i  = blockIdx.x * blockDim.x + threadIdx.x;
    long st = (long)gridDim.x * blockDim.x;
    for (; i < total; i += st) {
        int e   = base + (int)(i >> 2);
        int seg = (int)(i & 3);
        const float4* s = (const float4*)(T + (long)srcS[e] * DD + seg * 16);
        float* d = agg + (long)dstS[e] * DD + seg * 16;
#pragma unroll
        for (int q = 0; q < 4; ++q) {
            float4 v = s[q];
            atomicAdd(d + 4 * q + 0, v.x);
            atomicAdd(d + 4 * q + 1, v.y);
            atomicAdd(d + 4 * q + 2, v.z);
            atomicAdd(d + 4 * q + 3, v.w);
        }
    }
}

__global__ void relu_k(float* __restrict__ p, int n) {
    int i = blockIdx.x * blockDim.x + threadIdx.x;
    int st = gridDim.x * blockDim.x;
    for (; i < n; i += st) p[i] = fmaxf(p[i], 0.0f);
}

// column sums of h2 -> g[64]
__global__ void colsum_k(const float* __restrict__ h2, float* __restrict__ g, int nrows) {
    int tid = blockIdx.x * blockDim.x + threadIdx.x;
    int col = tid & 63;
    int rg  = tid >> 6;
    int nrg = (gridDim.x * blockDim.x) >> 6;
    float acc = 0.0f;
    for (int row = rg; row < nrows; row += nrg)
        acc += h2[(long)row * DD + col];
    atomicAdd(&g[col], acc);
}

// mean + 2-layer MLP head: out = sigmoid(relu(g*invN @ A1 + b1) @ A2 + b2)
__global__ void mlp_k(const float* __restrict__ g, const float* __restrict__ A1w,
                      const float* __restrict__ A1b, const float* __restrict__ A2w,
                      const float* __restrict__ A2b, float* __restrict__ out, float invN) {
    __shared__ float gm[64];
    __shared__ float hb[32];
    int t = threadIdx.x;
    gm[t] = g[t] * invN;
    __syncthreads();
    if (t < 32) {
        float s = A1b[t];
#pragma unroll 8
        for (int d = 0; d < 64; ++d) s += gm[d] * A1w[d * 32 + t];
        hb[t] = fmaxf(s, 0.0f);
    }
    __syncthreads();
    if (t < 10) {
        float s = A2b[t];
#pragma unroll 8
        for (int k = 0; k < 32; ++k) s += hb[k] * A2w[k * 10 + t];
        out[t] = 1.0f / (1.0f + expf(-s));
    }
}

// ---------------------------------------------------------------------------
extern "C" void kernel_launch(void* const* d_in, const int* in_sizes, int n_in,
                              void* d_out, int out_size, void* d_ws, size_t ws_size,
                              hipStream_t stream) {
    const float* h     = (const float*)d_in[0];
    const int*   src   = (const int*)d_in[1];
    const int*   dst   = (const int*)d_in[2];
    const int*   etype = (const int*)d_in[3];
    const float* W1    = (const float*)d_in[4];
    const float* W2    = (const float*)d_in[5];
    const float* A1w   = (const float*)d_in[6];
    const float* A1b   = (const float*)d_in[7];
    const float* A2w   = (const float*)d_in[8];
    const float* A2b   = (const float*)d_in[9];

    const int N = in_sizes[0] / DD;          // 100000
    const int E = in_sizes[1];               // 3200000
    const int R = in_sizes[4] / (DD * DD);   // 20
    const int ND = N * DD;                   // 6,400,000

    // workspace layout (float/int units, 4B each)
    float* ws   = (float*)d_ws;
    float* Tr   = ws;                        // [N*64] per-relation transform
    float* h1   = ws + (size_t)ND;           // [N*64] layer-1 agg / relu
    float* g    = ws + (size_t)2 * ND;       // [64]
    int* counts  = (int*)(ws + (size_t)2 * ND + 64);   // [R+1]
    int* offsets = counts + 32;                        // [R+1]
    int* cursor  = offsets + 32;                       // [R]
    int* srcS    = cursor + 32;                        // [E]
    int* dstS    = srcS + E;                           // [E]

    float* h2  = (float*)d_out;              // [N*64]
    float* att = h2 + (size_t)ND;            // [10]

    const int TB = 256;
    const int gridE   = 2048;                                  // grid-stride kernels
    const int gridZ   = 2048;
    const int gridXF  = (N / 16 * 32 + TB - 1) / TB;           // 1 wave / 16 rows

    // ---- counting sort of edges by relation (once) ----
    zeroi_k<<<1, 64, 0, stream>>>(counts, R + 1);
    hist_k<<<gridE, TB, 0, stream>>>(etype, counts, E);
    scan_k<<<1, 1, 0, stream>>>(counts, offsets, cursor, R);
    scatter_k<<<gridE, TB, 0, stream>>>(src, dst, etype, cursor, srcS, dstS, E);

    // ---- layer 1: agg into h1 ----
    zerof_k<<<gridZ, TB, 0, stream>>>(h1, ND);
    for (int r = 0; r < R; ++r) {
        transform_k<<<gridXF, TB, 0, stream>>>(h, W1 + (size_t)r * DD * DD, Tr, N);
        edge_agg_k<<<gridE, TB, 0, stream>>>(Tr, srcS, dstS, offsets, r, h1);
    }
    relu_k<<<gridZ, TB, 0, stream>>>(h1, ND);

    // ---- layer 2: agg directly into d_out (h2) ----
    zerof_k<<<gridZ, TB, 0, stream>>>(h2, ND);
    for (int r = 0; r < R; ++r) {
        transform_k<<<gridXF, TB, 0, stream>>>(h1, W2 + (size_t)r * DD * DD, Tr, N);
        edge_agg_k<<<gridE, TB, 0, stream>>>(Tr, srcS, dstS, offsets, r, h2);
    }

    // ---- mean over nodes + attributor MLP ----
    zerof_k<<<1, 64, 0, stream>>>(g, 64);
    colsum_k<<<1024, TB, 0, stream>>>(h2, g, N);
    mlp_k<<<1, 64, 0, stream>>>(g, A1w, A1b, A2w, A2b, att, 1.0f / (float)N);
}